// SlotAttention_84507776516664
// MI455X (gfx1250) — compile-verified
//
#include <hip/hip_runtime.h>
#include <hip/hip_bf16.h>

#define B_   16
#define N_   4096
#define D_   256
#define NS_  8
#define NP_  4
#define H_   256
#define EPS_ 1e-8f
#define SCALE_ 0.0625f   // 256^-0.5

typedef float v2f __attribute__((ext_vector_type(2)));
typedef float v8f __attribute__((ext_vector_type(8)));

#define CHUNK 256
#define LDA_P 264   // padded LDS row stride to dodge bank conflicts

// ---------------------------------------------------------------------------
// Per-row mean / rstd for LayerNorm of the big input (one wave32 per row)
// ---------------------------------------------------------------------------
__global__ void row_stats_kernel(const float* __restrict__ x,
                                 float* __restrict__ mean,
                                 float* __restrict__ rstd, int M) {
  int gid  = blockIdx.x * blockDim.x + threadIdx.x;
  int wave = gid >> 5;
  int lane = threadIdx.x & 31;
  if (wave >= M) return;
  const float* row = x + (long)wave * D_;
  float s = 0.f, ss = 0.f;
  for (int i = lane; i < D_; i += 32) { float t = row[i]; s += t; ss += t * t; }
  for (int off = 16; off > 0; off >>= 1) {
    s  += __shfl_xor(s,  off, 32);
    ss += __shfl_xor(ss, off, 32);
  }
  if (lane == 0) {
    float m   = s / (float)D_;
    float var = ss / (float)D_ - m * m;
    mean[wave] = m;
    rstd[wave] = rsqrtf(var + 1e-5f);
  }
}

// ---------------------------------------------------------------------------
// LayerNorm of small slot matrices (one wave per row of length D)
// ---------------------------------------------------------------------------
__global__ void ln_rows_kernel(const float* __restrict__ x, float* __restrict__ y,
                               const float* __restrict__ g, const float* __restrict__ b,
                               int M) {
  int gid  = blockIdx.x * blockDim.x + threadIdx.x;
  int wave = gid >> 5;
  int lane = threadIdx.x & 31;
  if (wave >= M) return;
  const float* row = x + (long)wave * D_;
  float s = 0.f, ss = 0.f;
  for (int i = lane; i < D_; i += 32) { float t = row[i]; s += t; ss += t * t; }
  for (int off = 16; off > 0; off >>= 1) {
    s  += __shfl_xor(s,  off, 32);
    ss += __shfl_xor(ss, off, 32);
  }
  float m  = s / (float)D_;
  float rs = rsqrtf(ss / (float)D_ - m * m + 1e-5f);
  float* out = y + (long)wave * D_;
  for (int i = lane; i < D_; i += 32)
    out[i] = (row[i] - m) * rs * g[i] + b[i];
}

// ---------------------------------------------------------------------------
// Fused LN(inputs) @ {Wk|Wv} + {bk|bv}  (blockIdx.z selects k vs v)
// Block = 512 threads = 16 waves; each wave owns one 16-col tile of the 256.
// ---------------------------------------------------------------------------
__global__ void gemm_ln_kv_kernel(const float* __restrict__ x,
                                  const float* __restrict__ mean,
                                  const float* __restrict__ rstd,
                                  const float* __restrict__ g,
                                  const float* __restrict__ bln,
                                  const float* __restrict__ Wk, const float* __restrict__ bk,
                                  const float* __restrict__ Wv, const float* __restrict__ bv,
                                  float* __restrict__ kout, float* __restrict__ vout) {
  __shared__ float ldsA[16 * LDA_P];
  const float* W    = blockIdx.z ? Wv : Wk;
  const float* bias = blockIdx.z ? bv : bk;
  float*       Cout = blockIdx.z ? vout : kout;

  long rowTile = (long)blockIdx.x * 16;
  for (int e = threadIdx.x; e < 16 * 256; e += blockDim.x) {
    int r = e >> 8, cc = e & 255;
    long grow = rowTile + r;
    float t = x[grow * D_ + cc];
    ldsA[r * LDA_P + cc] = (t - mean[grow]) * rstd[grow] * g[cc] + bln[cc];
  }
  __syncthreads();

  int waveId = threadIdx.x >> 5;
  int lane   = threadIdx.x & 31;
  int lo = lane & 15, hi = lane >> 4;
  int colBase = waveId * 16;

  v8f acc = {};
  for (int kk = 0; kk < 256; kk += 4) {
    int kA = kk + 2 * hi;
    v2f a, b;
    a.x = ldsA[lo * LDA_P + kA];
    a.y = ldsA[lo * LDA_P + kA + 1];
    b.x = W[(long)kA * D_ + colBase + lo];
    b.y = W[(long)(kA + 1) * D_ + colBase + lo];
    acc = __builtin_amdgcn_wmma_f32_16x16x4_f32(false, a, false, b,
                                                (short)0, acc, false, false);
  }
  int col = colBase + lo;
#pragma unroll
  for (int r = 0; r < 8; ++r) {
    long grow = rowTile + r + 8 * hi;
    Cout[grow * D_ + col] = acc[r] + bias[col];
  }
}

// ---------------------------------------------------------------------------
// Generic batched fp32 WMMA GEMM:  C = act(scale * A@B + bias) + addsrc
//   A: [M x K] row-major (stride lda), staged through LDS in 256-wide chunks,
//      zero padded for rows >= M.
//   B element (kk, col) at  B[kk*sK + col*sN]  (so sK=1,sN=ld gives B^T).
//   Grid: x = ceil(M/16), y covers 256 cols per block (16 waves), z = batch.
// ---------------------------------------------------------------------------
__global__ void gemm_f32_kernel(const float* __restrict__ A, int lda, long batchA,
                                const float* __restrict__ Bm, long sK, long sN, long batchB,
                                const float* __restrict__ bias,
                                const float* __restrict__ addsrc, long batchAdd,
                                float* __restrict__ C, int ldc, long batchC,
                                int M, int K, float scale, int relu) {
  __shared__ float ldsA[16 * LDA_P];
  int bz = blockIdx.z;
  const float* Ab   = A  + (long)bz * batchA;
  const float* Bb   = Bm + (long)bz * batchB;
  float*       Cb   = C  + (long)bz * batchC;
  const float* Addb = addsrc ? addsrc + (long)bz * batchAdd : nullptr;

  int rowTile = blockIdx.x * 16;
  int colBase = blockIdx.y * 256 + (int)(threadIdx.x >> 5) * 16;
  int lane = threadIdx.x & 31;
  int lo = lane & 15, hi = lane >> 4;

  v8f acc = {};
  for (int kc = 0; kc < K; kc += CHUNK) {
    for (int e = threadIdx.x; e < 16 * CHUNK; e += blockDim.x) {
      int r = e >> 8, cc = e & 255;
      int grow = rowTile + r;
      ldsA[r * LDA_P + cc] = (grow < M) ? Ab[(long)grow * lda + kc + cc] : 0.f;
    }
    __syncthreads();
    const float* Bk = Bb + (long)kc * sK + (long)colBase * sN;
    for (int kk = 0; kk < CHUNK; kk += 4) {
      int kA = kk + 2 * hi;
      v2f a, b;
      a.x = ldsA[lo * LDA_P + kA];
      a.y = ldsA[lo * LDA_P + kA + 1];
      b.x = Bk[(long)kA * sK + (long)lo * sN];
      b.y = Bk[(long)(kA + 1) * sK + (long)lo * sN];
      acc = __builtin_amdgcn_wmma_f32_16x16x4_f32(false, a, false, b,
                                                  (short)0, acc, false, false);
    }
    __syncthreads();
  }

  int col = colBase + lo;
#pragma unroll
  for (int r = 0; r < 8; ++r) {
    int grow = rowTile + r + 8 * hi;
    if (grow < M) {
      float cv = acc[r] * scale;
      if (bias) cv += bias[col];
      if (relu) cv = fmaxf(cv, 0.f);
      if (Addb) cv += Addb[(long)grow * ldc + col];
      Cb[(long)grow * ldc + col] = cv;
    }
  }
}

// ---------------------------------------------------------------------------
// Softmax over the NS (slot) axis for each (b, j); writes softmax + eps.
// ---------------------------------------------------------------------------
__global__ void sem_colsoftmax_kernel(const float* __restrict__ dots,
                                      float* __restrict__ attn, float eps) {
  int idx = blockIdx.x * blockDim.x + threadIdx.x;
  if (idx >= B_ * N_) return;
  int b = idx / N_, j = idx - b * N_;
  const float* dp = dots + (long)b * NS_ * N_ + j;
  float vals[NS_];
  float m = -1e30f;
#pragma unroll
  for (int i = 0; i < NS_; ++i) { vals[i] = dp[(long)i * N_]; m = fmaxf(m, vals[i]); }
  float s = 0.f;
#pragma unroll
  for (int i = 0; i < NS_; ++i) { vals[i] = expf(vals[i] - m); s += vals[i]; }
  float inv = 1.f / s;
  float* ap = attn + (long)b * NS_ * N_ + j;
#pragma unroll
  for (int i = 0; i < NS_; ++i) ap[(long)i * N_] = vals[i] * inv + eps;
}

// ---------------------------------------------------------------------------
// Softmax over the NP (instance) axis for each (b, ns, j); +EPS, * sem_w.
// ---------------------------------------------------------------------------
__global__ void inst_colsoftmax_kernel(const float* __restrict__ dots,
                                       const float* __restrict__ semw,
                                       float* __restrict__ attn) {
  int idx = blockIdx.x * blockDim.x + threadIdx.x;
  if (idx >= B_ * NS_ * N_) return;
  int j  = idx % N_;
  int t  = idx / N_;
  int ns = t % NS_;
  int b  = t / NS_;
  const float* dp = dots + ((long)(b * NS_ + ns) * NP_) * N_ + j;
  float vals[NP_];
  float m = -1e30f;
#pragma unroll
  for (int i = 0; i < NP_; ++i) { vals[i] = dp[(long)i * N_]; m = fmaxf(m, vals[i]); }
  float s = 0.f;
#pragma unroll
  for (int i = 0; i < NP_; ++i) { vals[i] = expf(vals[i] - m); s += vals[i]; }
  float inv = 1.f / s;
  float w   = semw[(long)(b * NS_ + ns) * N_ + j];
  float* ap = attn + ((long)(b * NS_ + ns) * NP_) * N_ + j;
#pragma unroll
  for (int i = 0; i < NP_; ++i) ap[(long)i * N_] = (vals[i] * inv + EPS_) * w;
}

// ---------------------------------------------------------------------------
// Normalize each attn row (length n) by its sum. One block per row.
// ---------------------------------------------------------------------------
__global__ void rownorm_kernel(float* __restrict__ attn, int n) {
  __shared__ float red[256];
  float* p = attn + (long)blockIdx.x * n;
  float s = 0.f;
  for (int i = threadIdx.x; i < n; i += blockDim.x) s += p[i];
  red[threadIdx.x] = s;
  __syncthreads();
  for (int off = blockDim.x >> 1; off > 0; off >>= 1) {
    if ((int)threadIdx.x < off) red[threadIdx.x] += red[threadIdx.x + off];
    __syncthreads();
  }
  float inv = 1.f / red[0];
  for (int i = threadIdx.x; i < n; i += blockDim.x) p[i] *= inv;
}

// ---------------------------------------------------------------------------
// Slot initializers
// ---------------------------------------------------------------------------
__global__ void sem_init_kernel(const float* __restrict__ mu, float* __restrict__ s) {
  int idx = blockIdx.x * blockDim.x + threadIdx.x;
  if (idx >= B_ * NS_ * D_) return;
  int d = idx % D_;
  int i = (idx / D_) % NS_;
  s[idx] = mu[i * D_ + d];
}

__global__ void inst_init_kernel(const float* __restrict__ mu,
                                 const float* __restrict__ sigma,
                                 const float* __restrict__ noise,
                                 float* __restrict__ s) {
  int idx = blockIdx.x * blockDim.x + threadIdx.x;
  if (idx >= B_ * NS_ * NP_ * D_) return;
  int d = idx % D_;
  int i = (idx / (D_ * NP_)) % NS_;
  s[idx] = mu[i * D_ + d] + expf(sigma[i * D_ + d]) * noise[idx];
}

// ---------------------------------------------------------------------------
// Host orchestration
// ---------------------------------------------------------------------------
extern "C" void kernel_launch(void* const* d_in, const int* in_sizes, int n_in,
                              void* d_out, int out_size, void* d_ws, size_t ws_size,
                              hipStream_t stream) {
  const float* inputs = (const float*)d_in[0];
  const float* noise  = (const float*)d_in[3];
  const float* ln_in_g = (const float*)d_in[4];
  const float* ln_in_b = (const float*)d_in[5];
  const float* ln_s_g  = (const float*)d_in[6];
  const float* ln_s_b  = (const float*)d_in[7];
  const float* ln_ff_g = (const float*)d_in[8];
  const float* ln_ff_b = (const float*)d_in[9];
  const float* mu    = (const float*)d_in[10];
  const float* sigma = (const float*)d_in[11];
  const float* Wq = (const float*)d_in[12]; const float* bq = (const float*)d_in[13];
  const float* Wk = (const float*)d_in[14]; const float* bk = (const float*)d_in[15];
  const float* Wv = (const float*)d_in[16]; const float* bv = (const float*)d_in[17];
  const float* W1 = (const float*)d_in[18]; const float* b1 = (const float*)d_in[19];
  const float* W2 = (const float*)d_in[20]; const float* b2 = (const float*)d_in[21];

  float* ws = (float*)d_ws;
  size_t off = 0;
  auto alloc = [&](size_t nelems) { float* p = ws + off; off += nelems; return p; };

  const int M = B_ * N_;
  float* kbuf  = alloc((size_t)M * D_);        // 64 MB
  float* vbuf  = alloc((size_t)M * D_);        // 64 MB
  float* meanb = alloc(M);
  float* rstdb = alloc(M);
  float* s     = alloc((size_t)B_ * NS_ * D_);
  float* sln   = alloc((size_t)B_ * NS_ * D_);
  float* qb    = alloc((size_t)B_ * NS_ * D_);
  float* hid   = alloc((size_t)B_ * NS_ * H_);
  float* dots  = alloc((size_t)B_ * NS_ * N_);
  float* attn  = alloc((size_t)B_ * NS_ * N_);
  float* semw  = alloc((size_t)B_ * NS_ * N_);
  float* is    = alloc((size_t)B_ * NS_ * NP_ * D_);
  float* isln  = alloc((size_t)B_ * NS_ * NP_ * D_);
  float* iq    = alloc((size_t)B_ * NS_ * NP_ * D_);
  float* ihid  = alloc((size_t)B_ * NS_ * NP_ * H_);
  float* idots = alloc((size_t)B_ * NS_ * NP_ * N_);
  float* iattn = alloc((size_t)B_ * NS_ * NP_ * N_);
  (void)ws_size; (void)in_sizes; (void)n_in; (void)out_size;

  // --- LN stats + fused k/v projection ---
  row_stats_kernel<<<M / 8, 256, 0, stream>>>(inputs, meanb, rstdb, M);
  gemm_ln_kv_kernel<<<dim3(M / 16, 1, 2), 512, 0, stream>>>(
      inputs, meanb, rstdb, ln_in_g, ln_in_b, Wk, bk, Wv, bv, kbuf, vbuf);

  // --- semantic slots ---
  sem_init_kernel<<<(B_ * NS_ * D_) / 256, 256, 0, stream>>>(mu, s);
  const int SM = B_ * NS_;  // 128 rows
  for (int t = 0; t < 3; ++t) {
    ln_rows_kernel<<<SM / 8, 256, 0, stream>>>(s, sln, ln_s_g, ln_s_b, SM);
    gemm_f32_kernel<<<dim3(SM / 16, 1, 1), 512, 0, stream>>>(
        sln, D_, 0, Wq, (long)D_, 1, 0, bq, nullptr, 0, qb, D_, 0, SM, D_, 1.f, 0);
    // dots[b,i,j] = scale * q[b,i,:] . k[b,j,:]
    gemm_f32_kernel<<<dim3(1, N_ / 256, B_), 512, 0, stream>>>(
        qb, D_, (long)NS_ * D_, kbuf, 1, (long)D_, (long)N_ * D_,
        nullptr, nullptr, 0, dots, N_, (long)NS_ * N_, NS_, D_, SCALE_, 0);
    sem_colsoftmax_kernel<<<(B_ * N_) / 256, 256, 0, stream>>>(dots, attn, EPS_);
    rownorm_kernel<<<SM, 256, 0, stream>>>(attn, N_);
    // slots += attn @ v
    gemm_f32_kernel<<<dim3(1, 1, B_), 512, 0, stream>>>(
        attn, N_, (long)NS_ * N_, vbuf, (long)D_, 1, (long)N_ * D_,
        nullptr, s, (long)NS_ * D_, s, D_, (long)NS_ * D_, NS_, N_, 1.f, 0);
    // MLP
    ln_rows_kernel<<<SM / 8, 256, 0, stream>>>(s, sln, ln_ff_g, ln_ff_b, SM);
    gemm_f32_kernel<<<dim3(SM / 16, 1, 1), 512, 0, stream>>>(
        sln, D_, 0, W1, (long)H_, 1, 0, b1, nullptr, 0, hid, H_, 0, SM, D_, 1.f, 1);
    gemm_f32_kernel<<<dim3(SM / 16, 1, 1), 512, 0, stream>>>(
        hid, H_, 0, W2, (long)D_, 1, 0, b2, s, 0, s, D_, 0, SM, H_, 1.f, 0);
  }
  // sem_w = softmax(sem_attn, axis=slots)   (no EPS)
  sem_colsoftmax_kernel<<<(B_ * N_) / 256, 256, 0, stream>>>(dots, semw, 0.f);

  // --- instance slots ---
  inst_init_kernel<<<(B_ * NS_ * NP_ * D_) / 256, 256, 0, stream>>>(mu, sigma, noise, is);
  const int IM = B_ * NS_ * NP_;       // 512 rows total
  const int IMB = NS_ * NP_;           // 32 rows per batch
  for (int t = 0; t < 3; ++t) {
    ln_rows_kernel<<<IM / 8, 256, 0, stream>>>(is, isln, ln_s_g, ln_s_b, IM);
    gemm_f32_kernel<<<dim3(IM / 16, 1, 1), 512, 0, stream>>>(
        isln, D_, 0, Wq, (long)D_, 1, 0, bq, nullptr, 0, iq, D_, 0, IM, D_, 1.f, 0);
    gemm_f32_kernel<<<dim3(IMB / 16, N_ / 256, B_), 512, 0, stream>>>(
        iq, D_, (long)IMB * D_, kbuf, 1, (long)D_, (long)N_ * D_,
        nullptr, nullptr, 0, idots, N_, (long)IMB * N_, IMB, D_, SCALE_, 0);
    inst_colsoftmax_kernel<<<(B_ * NS_ * N_) / 256, 256, 0, stream>>>(idots, semw, iattn);
    rownorm_kernel<<<IM, 256, 0, stream>>>(iattn, N_);
    gemm_f32_kernel<<<dim3(IMB / 16, 1, B_), 512, 0, stream>>>(
        iattn, N_, (long)IMB * N_, vbuf, (long)D_, 1, (long)N_ * D_,
        nullptr, is, (long)IMB * D_, is, D_, (long)IMB * D_, IMB, N_, 1.f, 0);
    ln_rows_kernel<<<IM / 8, 256, 0, stream>>>(is, isln, ln_ff_g, ln_ff_b, IM);
    gemm_f32_kernel<<<dim3(IM / 16, 1, 1), 512, 0, stream>>>(
        isln, D_, 0, W1, (long)H_, 1, 0, b1, nullptr, 0, ihid, H_, 0, IM, D_, 1.f, 1);
    gemm_f32_kernel<<<dim3(IM / 16, 1, 1), 512, 0, stream>>>(
        ihid, H_, 0, W2, (long)D_, 1, 0, b2, is, 0, is, D_, 0, IM, H_, 1.f, 0);
  }

  // --- assemble outputs: (sem_slots, sem_attn, inst_slots, inst_dots) ---
  float* out = (float*)d_out;
  size_t o0 = (size_t)B_ * NS_ * D_;          // 32768
  size_t o1 = (size_t)B_ * NS_ * N_;          // 524288
  size_t o2 = (size_t)B_ * NS_ * NP_ * D_;    // 131072
  size_t o3 = (size_t)B_ * NS_ * NP_ * N_;    // 2097152
  hipMemcpyAsync(out,                s,     o0 * sizeof(float), hipMemcpyDeviceToDevice, stream);
  hipMemcpyAsync(out + o0,           dots,  o1 * sizeof(float), hipMemcpyDeviceToDevice, stream);
  hipMemcpyAsync(out + o0 + o1,      is,    o2 * sizeof(float), hipMemcpyDeviceToDevice, stream);
  hipMemcpyAsync(out + o0 + o1 + o2, idots, o3 * sizeof(float), hipMemcpyDeviceToDevice, stream);
}